// RNN_py_43396349558991
// MI455X (gfx1250) — compile-verified
//
#include <hip/hip_runtime.h>
#include <math.h>

// Sizes from the reference
#define BATCH   32
#define SEQ     512
#define EMB     1024
#define HID     1024
#define LAYERS  4
#define MROWS   (SEQ * BATCH)        // 16384 GEMM rows, m = s*32 + b

typedef __attribute__((ext_vector_type(2))) float v2f;
typedef __attribute__((ext_vector_type(8))) float v8f;

// ---------------------------------------------------------------------------
// Phase 1: z[l, m, n] = x_row(m) . Wih[l][n][:] + bih[l][n] + bhh[l][n]
//   m = s*32 + b ;  A = x (16384 x 1024) ; B = Wih[l]^T (1024 x 1024)
// Workgroup: 256 threads = 8 waves. WG tile: 64(M) x 128(N).
// wave (wm 0..3, wn 0..1): 16(M) x 64(N) -> 4 accumulators of 16x16.
// K staged through LDS in chunks of 32 (both tiles row-major, stride 36).
// ---------------------------------------------------------------------------
#define KC   32
#define LDA  36   // 36 mod 64: m*36 mod 64 distinct for m=0..15 -> no bank conflicts

__global__ void __launch_bounds__(256)
rnn_phase1(const float* __restrict__ x, const float* __restrict__ Wih,
           const float* __restrict__ bih, const float* __restrict__ bhh,
           float* __restrict__ z)
{
    __shared__ float As[64 * LDA];    // 64 rows x KC
    __shared__ float Bs[128 * LDA];   // 128 rows (N-local) x KC   (Bs[n][k] = Wih[l][n][k])

    const int l    = blockIdx.z;
    const int m0   = blockIdx.y * 64;
    const int n0   = blockIdx.x * 128;
    const int tid  = threadIdx.x;
    const int wave = tid >> 5;
    const int lane = tid & 31;
    const int wm   = wave & 3;       // M subtile
    const int wn   = wave >> 2;      // N half (64 wide)
    const int lm   = lane & 15;
    const int hi   = lane >> 4;      // lane half selects K pair {0,1} vs {2,3}
    const int ksel = hi << 1;

    const float* Wl = Wih + (size_t)l * HID * EMB;

    v8f acc[4] = {};

    for (int kc = 0; kc < EMB; kc += KC) {
        __syncthreads();
        // stage A: 64 x 32 floats = 512 float4, 2 per thread
        #pragma unroll
        for (int i = 0; i < 2; ++i) {
            int idx = tid + i * 256;
            int r = idx >> 3, c4 = idx & 7;
            int m = m0 + r;
            int b = m & 31, s = m >> 5;
            float4 v = *(const float4*)(x + ((size_t)b * SEQ + s) * EMB + kc + c4 * 4);
            *(float4*)(As + r * LDA + c4 * 4) = v;
        }
        // stage B: 128 x 32 floats = 1024 float4, 4 per thread (plain Wih rows)
        #pragma unroll
        for (int i = 0; i < 4; ++i) {
            int idx = tid + i * 256;
            int r = idx >> 3, c4 = idx & 7;
            float4 v = *(const float4*)(Wl + (size_t)(n0 + r) * EMB + kc + c4 * 4);
            *(float4*)(Bs + r * LDA + c4 * 4) = v;
        }
        __syncthreads();

        const float* arow = As + (wm * 16 + lm) * LDA + ksel;
        #pragma unroll
        for (int k = 0; k < KC; k += 4) {
            v2f a = *(const v2f*)(arow + k);
            #pragma unroll
            for (int nt = 0; nt < 4; ++nt) {
                v2f b = *(const v2f*)(Bs + (wn * 64 + nt * 16 + lm) * LDA + ksel + k);
                acc[nt] = __builtin_amdgcn_wmma_f32_16x16x4_f32(
                    false, a, false, b, (short)0, acc[nt], false, false);
            }
        }
    }

    // epilogue: fold both biases, store z
    const int mbase = m0 + wm * 16 + hi * 8;
    #pragma unroll
    for (int nt = 0; nt < 4; ++nt) {
        int n = n0 + wn * 64 + nt * 16 + lm;
        float bias = bih[l * HID + n] + bhh[l * HID + n];
        float* zp = z + ((size_t)l * MROWS + mbase) * HID + n;
        #pragma unroll
        for (int r = 0; r < 8; ++r)
            zp[(size_t)r * HID] = acc[nt][r] + bias;
    }
}

// ---------------------------------------------------------------------------
// Phase 2: persistent recurrence. 128 WGs: layer = wg/32, g-slice = (wg%32)*32.
// WG: 128 threads = 4 waves -> 2x2 tiles of 16x16 covering (32 batch x 32 g).
// Whh slice (32x1024 f32) resident in LDS for the whole kernel;
// h(t-1) (32x1024 f32) re-staged into LDS each step.
// Per-layer software barrier (atomic counter in d_ws) between steps.
// ---------------------------------------------------------------------------
#define S2 1028   // 1028 mod 64 = 4 -> m*4 mod 64 distinct for m=0..15; rows 16B aligned

__device__ __forceinline__ void layer_barrier(unsigned* ctr, unsigned target)
{
    __syncthreads();
    if (threadIdx.x == 0) {
        __threadfence();                 // release our h stores to device scope
        atomicAdd(ctr, 1u);
        while (atomicAdd(ctr, 0u) < target)
            __builtin_amdgcn_s_sleep(2);
    }
    __syncthreads();
    __threadfence();                     // acquire: invalidate stale h lines
}

__global__ void __launch_bounds__(128)
rnn_phase2(const float* __restrict__ z, const float* __restrict__ Whh,
           float* __restrict__ hbuf, float* __restrict__ out, unsigned* bar)
{
    extern __shared__ float smem[];
    float* whh_s = smem;                 // 32 x S2
    float* h_s   = smem + 32 * S2;       // 32 x S2

    const int l   = blockIdx.x >> 5;
    const int g0  = (blockIdx.x & 31) * 32;
    const int tid = threadIdx.x;
    const int wave = tid >> 5;
    const int lane = tid & 31;
    const int wm   = wave & 1;           // batch half
    const int wn   = wave >> 1;          // g half
    const int lm   = lane & 15;
    const int hi   = lane >> 4;
    const int ksel = hi << 1;

    const float* Wl = Whh + (size_t)l * HID * HID;
    const float* zl = z + (size_t)l * MROWS * HID;
    float* hl = hbuf + (size_t)l * BATCH * HID;
    unsigned* ctr = bar + l;

    const int n  = g0 + wn * 16 + lm;    // this lane's output column
    const int b0 = wm * 16 + hi * 8;     // this lane's first batch row

    // ---- t = 0: h = tanh(z(0)) (h_prev == 0, skip the GEMM entirely) ----
    #pragma unroll
    for (int r = 0; r < 8; ++r) {
        int b = b0 + r;
        float v = tanhf(zl[(size_t)b * HID + n]);
        hl[(size_t)b * HID + n] = v;
        if (l == LAYERS - 1) out[((size_t)b * SEQ + 0) * HID + n] = v;
    }

    // ---- load resident Whh slice: Whh[l][g0+r][:] -> whh_s[r][k] ----
    #pragma unroll 4
    for (int i = 0; i < 64; ++i) {
        int idx = tid + i * 128;
        int r = idx >> 8, c4 = idx & 255;
        float4 v = *(const float4*)(Wl + (size_t)(g0 + r) * HID + c4 * 4);
        *(float4*)(whh_s + r * S2 + c4 * 4) = v;
    }

    layer_barrier(ctr, 32u);             // step 0 complete across the layer

    const float* arow = h_s   + (wm * 16 + lm) * S2 + ksel;
    const float* brow = whh_s + (wn * 16 + lm) * S2 + ksel;

    for (int t = 1; t < SEQ; ++t) {
        // stage full h(t-1) into LDS (coalesced float4, 64 per thread)
        #pragma unroll 4
        for (int i = 0; i < 64; ++i) {
            int idx = tid + i * 128;
            int r = idx >> 8, c4 = idx & 255;
            float4 v = *(const float4*)(hl + (size_t)r * HID + c4 * 4);
            *(float4*)(h_s + r * S2 + c4 * 4) = v;
        }
        __syncthreads();

        v8f acc = {};
        #pragma unroll 8
        for (int k = 0; k < HID; k += 4) {
            v2f a = *(const v2f*)(arow + k);
            v2f b = *(const v2f*)(brow + k);
            acc = __builtin_amdgcn_wmma_f32_16x16x4_f32(
                false, a, false, b, (short)0, acc, false, false);
        }

        const float* zt = zl + (size_t)t * BATCH * HID;
        #pragma unroll
        for (int r = 0; r < 8; ++r) {
            int b = b0 + r;
            float v = tanhf(zt[(size_t)b * HID + n] + acc[r]);
            hl[(size_t)b * HID + n] = v;
            if (l == LAYERS - 1) out[((size_t)b * SEQ + t) * HID + n] = v;
            if (t == SEQ - 1)
                out[(size_t)BATCH * SEQ * HID + ((size_t)(l * BATCH + b)) * HID + n] = v;
            if (t < SEQ - 1)
                __builtin_prefetch(zt + (size_t)BATCH * HID + (size_t)b * HID + n, 0, 1);
        }

        layer_barrier(ctr, 32u * (unsigned)(t + 1));
    }
}

// ---------------------------------------------------------------------------
extern "C" void kernel_launch(void* const* d_in, const int* in_sizes, int n_in,
                              void* d_out, int out_size, void* d_ws, size_t ws_size,
                              hipStream_t stream)
{
    (void)in_sizes; (void)n_in; (void)out_size; (void)ws_size;
    const float* x   = (const float*)d_in[0];
    const float* Wih = (const float*)d_in[1];
    const float* bih = (const float*)d_in[2];
    const float* Whh = (const float*)d_in[3];
    const float* bhh = (const float*)d_in[4];
    float* out = (float*)d_out;

    // workspace layout: z (4*16384*1024 f32 = 256 MB) | hbuf (4*32*1024 f32) | barriers
    float*    zbuf = (float*)d_ws;
    float*    hbuf = zbuf + (size_t)LAYERS * MROWS * HID;
    unsigned* bar  = (unsigned*)(hbuf + (size_t)LAYERS * BATCH * HID);

    hipMemsetAsync(bar, 0, LAYERS * sizeof(unsigned), stream);

    dim3 g1(HID / 128, MROWS / 64, LAYERS);   // (8, 256, 4)
    rnn_phase1<<<g1, 256, 0, stream>>>(x, Wih, bih, bhh, zbuf);

    size_t lds2 = (size_t)2 * 32 * S2 * sizeof(float);   // ~257 KB of the 320 KB WGP LDS
    rnn_phase2<<<dim3(LAYERS * 32), 128, lds2, stream>>>(zbuf, Whh, hbuf, out, bar);
}